// DWT_2D_87445534147354
// MI455X (gfx1250) — compile-verified
//
#include <hip/hip_runtime.h>

typedef __attribute__((ext_vector_type(2))) float v2f;
typedef __attribute__((ext_vector_type(8))) float v8f;

namespace {
constexpr int kInW          = 512;
constexpr int kOutW         = 256;
constexpr int kImgInElems   = 512 * 512;             // 262144
constexpr int kImgOutElems  = 256 * 256;             // 65536
constexpr int kNImg         = 16 * 64;               // 1024 (B*C)
constexpr int kGroupsPerRow = kOutW / 64;            // 4 groups of 64 out cols
constexpr int kGroupsPerImg = kOutW * kGroupsPerRow; // 1024
constexpr int kTotalGroups  = kNImg * kGroupsPerImg; // 1,048,576
constexpr int kPlaneOut     = kNImg * kImgOutElems;  // 67,108,864 per subband
} // namespace

// 2x2 Haar DWT via V_WMMA_F32_16X16X4_F32, 64 output pixels per iteration.
//   B (4x16): col N = output pixel; lanes 0-15 hold K=0,1 (float2, even input
//             row), lanes 16-31 hold K=2,3 (float2, odd input row) -> the raw
//             coalesced b64 loads ARE the B operand.
//   A (16x4): row M = Haar filter w[M&3] (ll,lh,hl,hh repeated 4x).
//   D: VGPR s = subband s twice: lanes 0-15 = row M=s (pixels 0-15 of this
//      WMMA), lanes 16-31 = row M=s+8 (same subband, pixels of lanes 16-31
//      of the *next* 16-pixel tile when fed from the second WMMA). So the
//      merge of two WMMAs into one full-wave store is a single v_cndmask.
__global__ __launch_bounds__(256)
void dwt2d_haar_wmma(const float* __restrict__ x,
                     const float* __restrict__ w,
                     float* __restrict__ o0, float* __restrict__ o1,
                     float* __restrict__ o2, float* __restrict__ o3)
{
    const int lane          = threadIdx.x & 31;
    const int waveInBlock   = threadIdx.x >> 5;
    const int wavesPerBlock = blockDim.x >> 5;
    const int waveId        = blockIdx.x * wavesPerBlock + waveInBlock;
    const int waveStride    = gridDim.x * wavesPerBlock;

    // A operand (16x4 f32): lanes 0-15 hold row M=lane K=0/1, lanes 16-31
    // hold row M=lane-16 K=2/3. w flat [4][2][2]: element K of filter k is
    // w[k*4 + K].
    const int sb   = lane & 3;
    const int koff = (lane >> 4) << 1;      // 0 or 2
    v2f A;
    A.x = w[sb * 4 + koff + 0];
    A.y = w[sb * 4 + koff + 1];

    const v8f cz = {0.f, 0.f, 0.f, 0.f, 0.f, 0.f, 0.f, 0.f};

    const int pcol = lane & 15;  // B column (pixel within 16-pixel tile)
    const int prow = lane >> 4;  // 0 -> even input row, 1 -> odd input row

    float* const outs[4] = {o0, o1, o2, o3};
    const v2f* xb = (const v2f*)x;

    for (int g = waveId; g < kTotalGroups; g += waveStride) {
        const int n   = g >> 10;           // image index (g / 1024)
        const int rem = g & 1023;
        const int oh  = rem >> 2;          // output row
        const int ow0 = (rem & 3) << 6;    // first of 64 output columns

        // Per-lane source: row (2*oh + prow), cols 2*(ow0 + pcol), as a
        // 32-bit float2 index (even float offset) -> saddr + voffset form.
        const int i2 = (n * kImgInElems
                        + ((oh << 1) + prow) * kInW
                        + (ow0 << 1) + (pcol << 1)) >> 1;

        const v2f b1 = __builtin_nontemporal_load(xb + i2 +  0);  // pixels  0-15
        const v2f b2 = __builtin_nontemporal_load(xb + i2 + 16);  // pixels 16-31
        const v2f b3 = __builtin_nontemporal_load(xb + i2 + 32);  // pixels 32-47
        const v2f b4 = __builtin_nontemporal_load(xb + i2 + 48);  // pixels 48-63

        // EXEC is all-1s here: loop bounds are wave-uniform, no divergence.
        v8f d1 = __builtin_amdgcn_wmma_f32_16x16x4_f32(
                     false, A, false, b1, (short)0, cz, false, false);
        v8f d2 = __builtin_amdgcn_wmma_f32_16x16x4_f32(
                     false, A, false, b2, (short)0, cz, false, false);
        v8f d3 = __builtin_amdgcn_wmma_f32_16x16x4_f32(
                     false, A, false, b3, (short)0, cz, false, false);
        v8f d4 = __builtin_amdgcn_wmma_f32_16x16x4_f32(
                     false, A, false, b4, (short)0, cz, false, false);

        // d_even[s] lanes 0-15 and d_odd[s] lanes 16-31 together hold
        // subband s for 32 consecutive output pixels -> one cndmask each.
        const int ov = n * kImgOutElems + oh * kOutW + ow0 + lane;
        #pragma unroll
        for (int s = 0; s < 4; ++s) {
            const float va = (lane < 16) ? d1[s] : d2[s];  // pixels  0-31
            const float vb = (lane < 16) ? d3[s] : d4[s];  // pixels 32-63
            __builtin_nontemporal_store(va, outs[s] + ov);
            __builtin_nontemporal_store(vb, outs[s] + ov + 32);
        }
    }
}

extern "C" void kernel_launch(void* const* d_in, const int* in_sizes, int n_in,
                              void* d_out, int out_size, void* d_ws, size_t ws_size,
                              hipStream_t stream) {
    (void)in_sizes; (void)n_in; (void)out_size; (void)d_ws; (void)ws_size;
    const float* x = (const float*)d_in[0];   // [16,64,512,512] fp32
    const float* w = (const float*)d_in[1];   // [4,2,2] fp32 Haar filters
    float* out = (float*)d_out;               // ll|lh|hl|hh concatenated

    // 4096 blocks x 256 threads = 32768 waves; 32 groups of 64 pixels per
    // wave, exact cover of 1,048,576 groups (no remainder, no divergence).
    dim3 grid(4096), block(256);
    hipLaunchKernelGGL(dwt2d_haar_wmma, grid, block, 0, stream,
                       x, w,
                       out + 0LL * kPlaneOut, out + 1LL * kPlaneOut,
                       out + 2LL * kPlaneOut, out + 3LL * kPlaneOut);
}